// GIN_21620865368392
// MI455X (gfx1250) — compile-verified
//
#include <hip/hip_runtime.h>
#include <hip/hip_bf16.h>

typedef __attribute__((ext_vector_type(2))) float v2f;
typedef __attribute__((ext_vector_type(8))) float v8f;

#define NN 100000
#define HH 128
#define EE 500000
#define GG 256
#define CHK 50000   /* edge chunk (multiple of 16; 500000/50000 = 10 chunks) */
#define KP 36       /* padded K for the 35-dim node-feature encoders */

static __device__ __forceinline__ float d_sigmoid(float z) {
    return __builtin_amdgcn_rcpf(1.f + __expf(-z));      // v_rcp_f32, no IEEE div
}
static __device__ __forceinline__ float d_silu(float z)  { return z * d_sigmoid(z); }
static __device__ __forceinline__ float d_lrelu(float z) { return z > 0.f ? z : 0.01f * z; }

static __device__ __forceinline__ v8f wmma4(v2f a, v2f b, v8f c) {
    return __builtin_amdgcn_wmma_f32_16x16x4_f32(false, a, false, b, (short)0, c, false, false);
}

static __device__ __forceinline__ float epi(float v, int outAct) {
    if      (outAct == 1) v = d_silu(v);
    else if (outAct == 2) v = d_lrelu(v);
    else if (outAct == 3) v = d_sigmoid(v);
    return v;
}

// ---------------------------------------------------------------------------
// fp32 WMMA GEMM (requires K%4==0, M%16==0, Ho%64==0 — guaranteed by padding).
// One wave -> 16x64 output tile: one A fragment feeds FOUR independent WMMA
// accumulator chains (4x A reuse). Inner loop: 5x global_load_b64 + 4x v_wmma,
// zero branches.
// ---------------------------------------------------------------------------
template<bool HASA2, bool INSIG>
__global__ __launch_bounds__(256) void
k_gemm4(const float* __restrict__ A, const float* __restrict__ A2,
        int M, int K, const float* __restrict__ W, int ldw, int kOff,
        const float* __restrict__ bias, float* __restrict__ Y,
        int Ho, int outAct, int accum)
{
    int  lane   = threadIdx.x & 31;
    long wid    = (((long)blockIdx.x * blockDim.x) + threadIdx.x) >> 5;
    int  tilesN = Ho >> 6;
    long tilesM = (long)(M >> 4);
    if (wid >= tilesM * tilesN) return;
    int tm = (int)(wid / tilesN), tn = (int)(wid % tilesN);
    int m0 = tm << 4, n0 = tn << 6;
    int l15 = lane & 15;
    int khalf = (lane >> 4) << 1;                  // lanes 0-15: K+{0,1}; 16-31: K+{2,3}

    const float* pa  = A  + (long)(m0 + l15) * K + khalf;
    const float* pa2 = A2 + (long)(m0 + l15) * K + khalf;   // dead unless HASA2
    long wb = (long)(n0 + l15) * ldw + kOff + khalf;
    const float* pw0 = W + wb;
    const float* pw1 = W + wb + 16L * ldw;
    const float* pw2 = W + wb + 32L * ldw;
    const float* pw3 = W + wb + 48L * ldw;

    v8f c0 = {0.f,0.f,0.f,0.f,0.f,0.f,0.f,0.f};
    v8f c1 = c0, c2 = c0, c3 = c0;
    #pragma unroll 2
    for (int k0 = 0; k0 < K; k0 += 4) {
        v2f av = *(const v2f*)pa;      pa  += 4;
        if (HASA2) { v2f a2 = *(const v2f*)pa2; pa2 += 4; av.x += a2.x; av.y += a2.y; }
        if (INSIG) { av.x = d_sigmoid(av.x); av.y = d_sigmoid(av.y); }
        v2f b0 = *(const v2f*)pw0;     pw0 += 4;
        v2f b1 = *(const v2f*)pw1;     pw1 += 4;
        v2f b2 = *(const v2f*)pw2;     pw2 += 4;
        v2f b3 = *(const v2f*)pw3;     pw3 += 4;
        c0 = wmma4(av, b0, c0);
        c1 = wmma4(av, b1, c1);
        c2 = wmma4(av, b2, c2);
        c3 = wmma4(av, b3, c3);
    }

    int rowBase = m0 + ((lane >> 4) << 3);         // C/D: vgpr r -> M=r / M=r+8
#define EPILOG(CREG, J)                                                         \
    {                                                                           \
        int col = n0 + (J) * 16 + l15;                                          \
        float bb = bias ? bias[col] : 0.f;                                      \
        _Pragma("unroll")                                                       \
        for (int r = 0; r < 8; ++r) {                                           \
            long idx = (long)(rowBase + r) * Ho + col;                          \
            float v = CREG[r] + bb;                                             \
            if (accum) v += Y[idx];                                             \
            Y[idx] = epi(v, outAct);                                            \
        }                                                                       \
    }
    EPILOG(c0, 0) EPILOG(c1, 1) EPILOG(c2, 2) EPILOG(c3, 3)
#undef EPILOG
}

// ---------------------------------------------------------------------------
// Elementwise / scatter / reduction kernels
// ---------------------------------------------------------------------------
__global__ void k_fill(float* p, float v, long n) {
    long t = (long)blockIdx.x * blockDim.x + threadIdx.x;
    long s = (long)gridDim.x * blockDim.x;
    for (; t < n; t += s) p[t] = v;
}

// zero-pad rows: src (rows x kin) -> dst (rows x kp), tail columns = 0
__global__ void k_pad(const float* __restrict__ src, float* __restrict__ dst,
                      long rows, int kin, int kp) {
    long t = (long)blockIdx.x * blockDim.x + threadIdx.x;
    if (t >= rows * kp) return;
    long r = t / kp; int c = (int)(t - r * kp);
    dst[t] = (c < kin) ? src[r * kin + c] : 0.f;
}

__global__ void k_edge_dist(const int* __restrict__ ei, const float* __restrict__ pos,
                            float* __restrict__ d, int E) {
    long e = (long)blockIdx.x * blockDim.x + threadIdx.x;
    if (e >= E) return;
    int a = ei[e], b = ei[E + e];
    float dx = pos[3*a]   - pos[3*b];
    float dy = pos[3*a+1] - pos[3*b+1];
    float dz = pos[3*a+2] - pos[3*b+2];
    d[e] = sqrtf(dx*dx + dy*dy + dz*dz);
}

__global__ void k_rbf(const float* __restrict__ d, int eOff, int cnt, float* __restrict__ R) {
    long t = (long)blockIdx.x * blockDim.x + threadIdx.x;
    if (t >= (long)cnt * 16) return;
    int e = (int)(t >> 4), j = (int)(t & 15);
    float mu = 0.4f * (float)j;                      // linspace(0,6,16)
    float z  = (d[eOff + e] - mu) * (1.f / 0.375f);  // sig = 6/16
    R[t] = __expf(-z * z);
}

__global__ void k_rownorm(float* __restrict__ x, int M) {   // x <- 1.8 * x / max(|x|,eps)
    int  lane = threadIdx.x & 31;
    long w = (((long)blockIdx.x * blockDim.x) + threadIdx.x) >> 5;
    if (w >= M) return;
    float* row = x + w * (long)HH;
    float v[4]; float s = 0.f;
    #pragma unroll
    for (int i = 0; i < 4; ++i) { v[i] = row[lane + 32*i]; s += v[i]*v[i]; }
    #pragma unroll
    for (int o = 16; o > 0; o >>= 1) s += __shfl_xor(s, o, 32);
    float sc = 1.8f / fmaxf(sqrtf(s), 1e-12f);
    #pragma unroll
    for (int i = 0; i < 4; ++i) row[lane + 32*i] = v[i] * sc;
}

__global__ void k_deg(const int* __restrict__ ei, const float* __restrict__ w,
                      float* __restrict__ deg, int E) {
    long e = (long)blockIdx.x * blockDim.x + threadIdx.x;
    if (e >= E) return;
    atomicAdd(&deg[ei[E + e]], w[e]);
}

__global__ void k_appnp_scatter(const int* __restrict__ ei, const float* __restrict__ w,
                                const float* __restrict__ deg, const float* __restrict__ x,
                                float* __restrict__ prop, int E) {
    long t = (long)blockIdx.x * blockDim.x + threadIdx.x;
    if (t >= (long)E * 32) return;
    int e  = (int)(t >> 5);
    int f0 = (int)(t & 31) << 2;
    int r = ei[e], c = ei[E + e];
    float coef = rsqrtf(fmaxf(deg[r], 1e-12f)) * w[e] * rsqrtf(fmaxf(deg[c], 1e-12f));
    const float* xr = x    + (long)r * HH + f0;
    float*       pr = prop + (long)c * HH + f0;
    #pragma unroll
    for (int q = 0; q < 4; ++q) atomicAdd(pr + q, coef * xr[q]);
}

__global__ void k_appnp_fin(const float* __restrict__ xp, const float* __restrict__ deg,
                            float* __restrict__ prop, int M) {
    long t = (long)blockIdx.x * blockDim.x + threadIdx.x;
    if (t >= (long)M * HH) return;
    int i = (int)(t >> 7);
    prop[t] = 0.1f * xp[t] + 0.9f * (prop[t] + xp[t] * __builtin_amdgcn_rcpf(deg[i]));
}

__global__ void k_pool(const float* __restrict__ x, const int* __restrict__ batch,
                       float* __restrict__ pool, int M) {
    long t = (long)blockIdx.x * blockDim.x + threadIdx.x;
    if (t >= (long)M * 32) return;
    int i  = (int)(t >> 5);
    int f0 = (int)(t & 31) << 2;
    const float* xr = x + (long)i * HH + f0;
    float* pr = pool + (long)batch[i] * HH + f0;
    #pragma unroll
    for (int q = 0; q < 4; ++q) atomicAdd(pr + q, xr[q]);
}

__global__ void k_scatter_rows(const float* __restrict__ rows, const int* __restrict__ ei,
                               float* __restrict__ out, int E) {
    long t = (long)blockIdx.x * blockDim.x + threadIdx.x;
    if (t >= (long)E * 32) return;
    int e  = (int)(t >> 5);
    int f0 = (int)(t & 31) << 2;
    int s = ei[e], d = ei[E + e];
    const float* sr = rows + (long)s * HH + f0;
    float*       dr = out  + (long)d * HH + f0;
    #pragma unroll
    for (int q = 0; q < 4; ++q) atomicAdd(dr + q, sr[q]);
}

// Fused GINE edge op: ea = silu(U1[s]+U2[d]+E3[e]+b); agg[d] += relu(xx[s]+ea)
__global__ void k_gine_edge(const int* __restrict__ ei, int E, int eOff, int cnt,
                            const float* __restrict__ U1, const float* __restrict__ U2,
                            const float* __restrict__ E3, const float* __restrict__ bupd,
                            const float* __restrict__ xx, float* __restrict__ agg) {
    long t = (long)blockIdx.x * blockDim.x + threadIdx.x;
    if (t >= (long)cnt * 32) return;
    int i  = (int)(t >> 5);
    int f0 = (int)(t & 31) << 2;
    long e = (long)eOff + i;
    int s = ei[e], d = ei[(long)E + e];
    const float* u1 = U1 + (long)s * HH;
    const float* u2 = U2 + (long)d * HH;
    const float* e3 = E3 + (long)i * HH;
    const float* xs = xx + (long)s * HH;
    float*       ag = agg + (long)d * HH;
    #pragma unroll
    for (int q = 0; q < 4; ++q) {
        int f = f0 + q;
        float v = d_silu(u1[f] + u2[f] + e3[f] + bupd[f]);
        float r = xs[f] + v; r = r > 0.f ? r : 0.f;
        atomicAdd(ag + f, r);
    }
}

__global__ void k_bn_stats(const float* __restrict__ x, int M, int C, float* __restrict__ sums) {
    int c = threadIdx.x;                         // blockDim.x == C
    float s = 0.f, q = 0.f;
    for (int r = blockIdx.x; r < M; r += gridDim.x) {
        float v = x[(long)r * C + c]; s += v; q += v * v;
    }
    atomicAdd(&sums[c], s);
    atomicAdd(&sums[C + c], q);
}

__global__ void k_bn_apply(float* __restrict__ x, int M, int C, const float* __restrict__ sums,
                           const float* __restrict__ g, const float* __restrict__ b) {
    long t = (long)blockIdx.x * blockDim.x + threadIdx.x;
    if (t >= (long)M * C) return;
    int c = (int)(t % C);
    float invM = 1.f / (float)M;
    float m   = sums[c] * invM;
    float var = sums[C + c] * invM - m * m;
    x[t] = g[c] * (x[t] - m) * rsqrtf(var + 1e-5f) + b[c];
}

__global__ void k_add2_silu(const float* a, const float* b, float* o, long n) {
    long t = (long)blockIdx.x * blockDim.x + threadIdx.x;
    if (t < n) o[t] = d_silu(a[t] + b[t]);
}

__global__ void k_add3(const float* a, const float* b, const float* c, float* o, long n) {
    long t = (long)blockIdx.x * blockDim.x + threadIdx.x;
    if (t < n) o[t] = a[t] + b[t] + c[t];
}

__global__ void k_out(const float* __restrict__ pooled, const float* __restrict__ xg,
                      const float* __restrict__ W, const float* __restrict__ b,
                      float* __restrict__ out) {
    int g = blockIdx.x * blockDim.x + threadIdx.x;
    if (g >= GG) return;
    float s = b[0];
    for (int f = 0; f < HH; ++f)
        s += W[f] * pooled[(long)g * HH + f] + W[HH + f] * xg[(long)g * HH + f];
    out[g] = s;
}

// ---------------------------------------------------------------------------
// Host-side orchestration
// ---------------------------------------------------------------------------
static inline int cdiv(long a, long b) { return (int)((a + b - 1) / b); }

static void gemm(hipStream_t st, const float* A, const float* A2, int inAct,
                 int M, int K, const float* W, int ldw, int kOff,
                 const float* bias, float* Y, int Ho, int outAct, int accum) {
    long waves = (long)(M >> 4) * (Ho >> 6);    // 16x64 tiles
    dim3 grid(cdiv(waves, 8)), blk(256);
    if (A2)
        hipLaunchKernelGGL((k_gemm4<true,  false>), grid, blk, 0, st, A, A2, M, K, W, ldw, kOff, bias, Y, Ho, outAct, accum);
    else if (inAct == 1)
        hipLaunchKernelGGL((k_gemm4<false, true >), grid, blk, 0, st, A, A,  M, K, W, ldw, kOff, bias, Y, Ho, outAct, accum);
    else
        hipLaunchKernelGGL((k_gemm4<false, false>), grid, blk, 0, st, A, A,  M, K, W, ldw, kOff, bias, Y, Ho, outAct, accum);
}

static void fillf(hipStream_t st, float* p, float v, long n) {
    hipLaunchKernelGGL(k_fill, dim3(4096), dim3(256), 0, st, p, v, n);
}

static void batchnorm(hipStream_t st, float* x, int M, int C, float* sums,
                      const float* g, const float* b) {
    fillf(st, sums, 0.f, 2L * C);
    hipLaunchKernelGGL(k_bn_stats, dim3(720), dim3(C), 0, st, x, M, C, sums);
    hipLaunchKernelGGL(k_bn_apply, dim3(cdiv((long)M * C, 256)), dim3(256), 0, st, x, M, C, sums, g, b);
}

// One GINE+DGNN branch (shared by inter/intra edge sets)
static void run_branch(hipStream_t st, const int* ei,
                       const float* pos, const float* xx, const float* U1, const float* U2,
                       const float* eupdW, const float* eupdB,
                       const float* eattrW, const float* eattrB,
                       const float* ginW, const float* ginB, const float* ginG, const float* ginBb,
                       const float* const* dg,     // 12: {selfW,selfB,neighW,neighB} x 3
                       const float* linW, const float* linB,
                       float* outBuf,
                       float* dE, float* rbfCh, float* zCh, float* e3Ch,
                       float* agg, float* gineBuf, float* hs, float* hn,
                       float* hA, float* hB, float* sums) {
    const long NHf = (long)NN * HH;

    hipLaunchKernelGGL(k_edge_dist, dim3(cdiv(EE, 256)), dim3(256), 0, st, ei, pos, dE, EE);
    fillf(st, agg, 0.f, NHf);

    for (int eOff = 0; eOff < EE; eOff += CHK) {
        int cnt = (EE - eOff) < CHK ? (EE - eOff) : CHK;
        hipLaunchKernelGGL(k_rbf, dim3(cdiv((long)cnt * 16, 256)), dim3(256), 0, st, dE, eOff, cnt, rbfCh);
        // Z = rbf @ Wattr^T + battr   (pre-sigmoid)
        gemm(st, rbfCh, nullptr, 0, cnt, 16, eattrW, 16, 0, eattrB, zCh, HH, 0, 0);
        // E3 = sigmoid(Z) @ W3^T      (W3 = edge_upd.W[:, 256:384])
        gemm(st, zCh, nullptr, 1, cnt, HH, eupdW, 3 * HH, 2 * HH, nullptr, e3Ch, HH, 0, 0);
        hipLaunchKernelGGL(k_gine_edge, dim3(cdiv((long)cnt * 32, 256)), dim3(256), 0, st,
                           ei, EE, eOff, cnt, U1, U2, e3Ch, eupdB, xx, agg);
    }

    // GINE node update: bn(lrelu((xx + agg) @ Wgin^T + bgin))
    gemm(st, xx, agg, 0, NN, HH, ginW, HH, 0, ginB, gineBuf, HH, 2, 0);
    batchnorm(st, gineBuf, NN, HH, sums, ginG, ginBb);

    // DGNN: 3 layers of silu(self(h) + seg(neigh(h)[src], dst))
    const float* hcur = xx;
    float* ping[3] = { hA, hB, hA };
    for (int l = 0; l < 3; ++l) {
        gemm(st, hcur, nullptr, 0, NN, HH, dg[4*l + 0], HH, 0, dg[4*l + 1], hs, HH, 0, 0);
        gemm(st, hcur, nullptr, 0, NN, HH, dg[4*l + 2], HH, 0, dg[4*l + 3], hn, HH, 0, 0);
        fillf(st, agg, 0.f, NHf);
        hipLaunchKernelGGL(k_scatter_rows, dim3(cdiv((long)EE * 32, 256)), dim3(256), 0, st, hn, ei, agg, EE);
        hipLaunchKernelGGL(k_add2_silu, dim3(cdiv(NHf, 256)), dim3(256), 0, st, hs, agg, ping[l], NHf);
        hcur = ping[l];
    }

    // out = silu(concat[gine, dgnn] @ linW^T + linB), decomposed over K-offsets
    gemm(st, gineBuf, nullptr, 0, NN, HH, linW, 2 * HH, 0,  linB,    outBuf, HH, 0, 0);
    gemm(st, hA,      nullptr, 0, NN, HH, linW, 2 * HH, HH, nullptr, outBuf, HH, 1, 1);
}

extern "C" void kernel_launch(void* const* d_in, const int* in_sizes, int n_in,
                              void* d_out, int out_size, void* d_ws, size_t ws_size,
                              hipStream_t stream) {
    const float* x        = (const float*)d_in[0];
    const float* pos      = (const float*)d_in[1];
    const int*   ei_inter = (const int*)d_in[2];
    const int*   ei_intra = (const int*)d_in[3];
    const int*   ei_aug   = (const int*)d_in[4];
    const int*   batch    = (const int*)d_in[5];
    // params flattened in setup_inputs() insertion order:
    const float* encW  = (const float*)d_in[6];   const float* encB  = (const float*)d_in[7];
    const float* lnW   = (const float*)d_in[8];   const float* lnB   = (const float*)d_in[9];
    const float* empW  = (const float*)d_in[10];  const float* empB  = (const float*)d_in[11];
    const float* ebnG  = (const float*)d_in[12];  const float* ebnB  = (const float*)d_in[13];
    const float* eatW  = (const float*)d_in[14];  const float* eatB  = (const float*)d_in[15];
    const float* eupW  = (const float*)d_in[16];  const float* eupB  = (const float*)d_in[17];
    const float* g1W   = (const float*)d_in[18];  const float* g1B   = (const float*)d_in[19];
    const float* g1G   = (const float*)d_in[20];  const float* g1Bb  = (const float*)d_in[21];
    const float* g3W   = (const float*)d_in[22];  const float* g3B   = (const float*)d_in[23];
    const float* g3G   = (const float*)d_in[24];  const float* g3Bb  = (const float*)d_in[25];
    const float* g4W   = (const float*)d_in[26];  const float* g4B   = (const float*)d_in[27];
    const float* g4G   = (const float*)d_in[28];  const float* g4Bb  = (const float*)d_in[29];
    const float* dg1[12]; for (int i = 0; i < 12; ++i) dg1[i] = (const float*)d_in[30 + i];
    const float* dg3[12]; for (int i = 0; i < 12; ++i) dg3[i] = (const float*)d_in[42 + i];
    const float* l1W   = (const float*)d_in[54];  const float* l1B   = (const float*)d_in[55];
    const float* l3W   = (const float*)d_in[56];  const float* l3B   = (const float*)d_in[57];
    const float* f0W   = (const float*)d_in[58];  const float* f0B   = (const float*)d_in[59];
    const float* f0G   = (const float*)d_in[60];  const float* f0Bb  = (const float*)d_in[61];
    const float* f1W   = (const float*)d_in[62];  const float* f1B   = (const float*)d_in[63];
    const float* f1G   = (const float*)d_in[64];  const float* f1Bb  = (const float*)d_in[65];
    const float* f2W   = (const float*)d_in[66];  const float* f2B   = (const float*)d_in[67];
    const float* f2G   = (const float*)d_in[68];  const float* f2Bb  = (const float*)d_in[69];
    const float* ffW   = (const float*)d_in[70];  const float* ffB   = (const float*)d_in[71];
    const float* loW   = (const float*)d_in[72];  const float* loB   = (const float*)d_in[73];

    // ---- workspace layout ----
    const long NHf = (long)NN * HH;
    float* ws = (float*)d_ws;
    float* b0  = ws;                 // xe -> x_psc -> xx
    float* b1  = b0  + NHf;          // x_inter0 -> U1     (fc: bT part 1)
    float* b2  = b1  + NHf;          // x_raw    -> U2     (fc: bT part 2)
    float* b3  = b2  + NHf;          // agg/magg           (fc: bT2 part 1)
    float* b4  = b3  + NHf;          // gine / x_mask      (fc: bT2 part 2)
    float* b5  = b4  + NHf;          // dgnn hs  -> fc L3 out
    float* b6  = b5  + NHf;          // dgnn hn  -> fc final out
    float* b7  = b6  + NHf;          // dgnn pingA / h
    float* b8  = b7  + NHf;          // dgnn pingB
    float* b9  = b8  + NHf;          // x_inter
    float* b10 = b9  + NHf;          // x_intra
    float* dE    = b10   + NHf;                 // E
    float* rbfCh = dE    + EE;                  // CHK*16
    float* zCh   = rbfCh + (long)CHK * 16;      // CHK*128
    float* e3Ch  = zCh   + (long)CHK * HH;      // CHK*128
    float* deg   = e3Ch  + (long)CHK * HH;      // N
    float* sums  = deg   + NN;                  // 512
    float* pooled= sums  + 512;                 // G*128
    float* xg    = pooled+ (long)GG * HH;       // G*128
    float* xP    = xg    + (long)GG * HH;       // N*36  (padded node features)
    float* encWP = xP    + (long)NN * KP;       // 128*36
    float* lnWP  = encWP + (long)HH * KP;       // 128*36
    float* bT  = b1;   // N x 256 (b1+b2 contiguous, dead in fc phase)
    float* bT2 = b3;   // N x 256 (b3+b4 contiguous, dead in fc phase)

    // 0. zero-pad node features and the two 35-dim weight matrices to K=36
    hipLaunchKernelGGL(k_pad, dim3(cdiv((long)NN * KP, 256)), dim3(256), 0, stream, x,   xP,    NN, 35, KP);
    hipLaunchKernelGGL(k_pad, dim3(cdiv((long)HH * KP, 256)), dim3(256), 0, stream, encW, encWP, HH, 35, KP);
    hipLaunchKernelGGL(k_pad, dim3(cdiv((long)HH * KP, 256)), dim3(256), 0, stream, lnW,  lnWP,  HH, 35, KP);

    // 1. xe = silu(enc_lin(x)); x_psc = 1.8 * normalize(xe)
    gemm(stream, xP, nullptr, 0, NN, KP, encWP, KP, 0, encB, b0, HH, 1, 0);
    hipLaunchKernelGGL(k_rownorm, dim3(cdiv((long)NN * 32, 256)), dim3(256), 0, stream, b0, NN);

    // 2. appnp over ei_inter with distance edge weights -> x_inter0 (b1)
    hipLaunchKernelGGL(k_edge_dist, dim3(cdiv(EE, 256)), dim3(256), 0, stream, ei_inter, pos, dE, EE);
    fillf(stream, deg, 1.0f, NN);   // self-loop weight
    hipLaunchKernelGGL(k_deg, dim3(cdiv(EE, 256)), dim3(256), 0, stream, ei_inter, dE, deg, EE);
    fillf(stream, b1, 0.f, NHf);
    hipLaunchKernelGGL(k_appnp_scatter, dim3(cdiv((long)EE * 32, 256)), dim3(256), 0, stream,
                       ei_inter, dE, deg, b0, b1, EE);
    hipLaunchKernelGGL(k_appnp_fin, dim3(cdiv(NHf, 256)), dim3(256), 0, stream, b0, deg, b1, NN);

    // 3. xg = global_add_pool(x_inter0)
    fillf(stream, xg, 0.f, (long)GG * HH);
    hipLaunchKernelGGL(k_pool, dim3(cdiv((long)NN * 32, 256)), dim3(256), 0, stream, b1, batch, xg, NN);

    // 4. x_raw = silu(lin_node(x))
    gemm(stream, xP, nullptr, 0, NN, KP, lnWP, KP, 0, lnB, b2, HH, 1, 0);

    // 5. xx = bn(lrelu(enc_mlp(x_inter0 + x_raw)))   -> b0
    gemm(stream, b1, b2, 0, NN, HH, empW, HH, 0, empB, b0, HH, 2, 0);
    batchnorm(stream, b0, NN, HH, sums, ebnG, ebnB);

    // 6. U1 = xx @ W_upd[:, :128]^T ; U2 = xx @ W_upd[:, 128:256]^T
    gemm(stream, b0, nullptr, 0, NN, HH, eupW, 3 * HH, 0,  nullptr, b1, HH, 0, 0);
    gemm(stream, b0, nullptr, 0, NN, HH, eupW, 3 * HH, HH, nullptr, b2, HH, 0, 0);

    // 7. inter branch -> x_inter (b9); intra branch -> x_intra (b10)
    run_branch(stream, ei_inter, pos, b0, b1, b2, eupW, eupB, eatW, eatB,
               g1W, g1B, g1G, g1Bb, dg1, l1W, l1B, b9,
               dE, rbfCh, zCh, e3Ch, b3, b4, b5, b6, b7, b8, sums);
    run_branch(stream, ei_intra, pos, b0, b1, b2, eupW, eupB, eatW, eatB,
               g3W, g3B, g3G, g3Bb, dg3, l3W, l3B, b10,
               dE, rbfCh, zCh, e3Ch, b3, b4, b5, b6, b7, b8, sums);

    // 8. x_mask = gin4(xx, ei_aug) -> b4
    fillf(stream, b3, 0.f, NHf);
    hipLaunchKernelGGL(k_scatter_rows, dim3(cdiv((long)EE * 32, 256)), dim3(256), 0, stream, b0, ei_aug, b3, EE);
    gemm(stream, b0, b3, 0, NN, HH, g4W, HH, 0, g4B, b4, HH, 2, 0);
    batchnorm(stream, b4, NN, HH, sums, g4G, g4Bb);

    // 9. h = x_inter + x_intra + x_mask -> b7
    hipLaunchKernelGGL(k_add3, dim3(cdiv(NHf, 256)), dim3(256), 0, stream, b9, b10, b4, b7, NHf);

    // 10. fc stack
    gemm(stream, b7,  nullptr, 0, NN, HH,      f0W, HH,      0, f0B, bT,  2 * HH, 2, 0);
    batchnorm(stream, bT, NN, 2 * HH, sums, f0G, f0Bb);
    gemm(stream, bT,  nullptr, 0, NN, 2 * HH,  f1W, 2 * HH,  0, f1B, bT2, 2 * HH, 2, 0);
    batchnorm(stream, bT2, NN, 2 * HH, sums, f1G, f1Bb);
    gemm(stream, bT2, nullptr, 0, NN, 2 * HH,  f2W, 2 * HH,  0, f2B, b5,  HH,     2, 0);
    batchnorm(stream, b5, NN, HH, sums, f2G, f2Bb);
    gemm(stream, b5,  nullptr, 0, NN, HH,      ffW, HH,      0, ffB, b6,  HH,     0, 0);

    // 11. pooled = global_add_pool(h_final); out = lin_out([pooled, xg])
    fillf(stream, pooled, 0.f, (long)GG * HH);
    hipLaunchKernelGGL(k_pool, dim3(cdiv((long)NN * 32, 256)), dim3(256), 0, stream, b6, batch, pooled, NN);
    hipLaunchKernelGGL(k_out, dim3(1), dim3(GG), 0, stream, pooled, xg, loW, loB, (float*)d_out);
}